// GCN_24146306138775
// MI455X (gfx1250) — compile-verified
//
#include <hip/hip_runtime.h>

typedef __attribute__((ext_vector_type(2))) float v2f;
typedef __attribute__((ext_vector_type(8))) float v8f;

#define IN_C  65
#define HID_C 64
#define PAD_K 68   // IN_C padded up to a multiple of 4; also LDS row stride (avoids bank conflicts)

// ---------------------------------------------------------------------------
// Hin[n][0:65] = x[n][0:65]  (self term, eps=0), Hin[n][65:68] = 0 (K padding)
// ---------------------------------------------------------------------------
__global__ void gin_init_hin(const float* __restrict__ x,
                             float* __restrict__ hin, int n_nodes) {
    int idx = blockIdx.x * blockDim.x + threadIdx.x;
    int total = n_nodes * PAD_K;
    if (idx >= total) return;
    int n = idx / PAD_K;
    int c = idx - n * PAD_K;
    hin[idx] = (c < IN_C) ? x[n * IN_C + c] : 0.0f;
}

// ---------------------------------------------------------------------------
// w1p[68][64]: rows 0..64 = w1, rows 65..67 = 0
// ---------------------------------------------------------------------------
__global__ void gin_pad_w1(const float* __restrict__ w1, float* __restrict__ w1p) {
    int idx = blockIdx.x * blockDim.x + threadIdx.x;
    if (idx >= PAD_K * HID_C) return;
    int r = idx / HID_C;
    int c = idx - r * HID_C;
    w1p[idx] = (r < IN_C) ? w1[r * HID_C + c] : 0.0f;
}

// ---------------------------------------------------------------------------
// Edge scatter: one wave32 per edge, lanes cover channels.
// agg(dst) += x(src); both buffers are L2-resident (26/27 MB vs 192 MB L2).
// ---------------------------------------------------------------------------
__global__ void gin_scatter(const float* __restrict__ x,
                            const long long* __restrict__ edge_index,
                            float* __restrict__ hin, int n_edges) {
    int warp = (blockIdx.x * blockDim.x + threadIdx.x) >> 5;
    int lane = threadIdx.x & 31;
    if (warp >= n_edges) return;
    long long src = edge_index[warp];                // row 0: message sources
    long long dst = edge_index[(long long)n_edges + warp]; // row 1: targets
    const float* xs = x + src * IN_C;
    float* hd = hin + dst * PAD_K;
    for (int c = lane; c < IN_C; c += 32)
        atomicAdd(&hd[c], xs[c]);
}

// ---------------------------------------------------------------------------
// Fused MLP: out = relu(Hin @ w1 + b1) @ w2 + b2 using V_WMMA_F32_16X16X4_F32.
// One block = 4 waves = one 16-node row tile; wave w owns output columns
// [16w, 16w+16). h tile lives in LDS between the two GEMMs.
// ---------------------------------------------------------------------------
__global__ __launch_bounds__(128)
void gin_mlp_wmma(const float* __restrict__ hin,
                  const float* __restrict__ w1p,
                  const float* __restrict__ b1,
                  const float* __restrict__ w2,
                  const float* __restrict__ b2,
                  float* __restrict__ out, int n_nodes) {
    __shared__ float lds_in[16 * PAD_K];  // 16 x 68 input tile
    __shared__ float lds_h[16 * PAD_K];   // 16 x 64 hidden tile (stride 68)

    const int m0   = blockIdx.x * 16;
    const int tid  = threadIdx.x;
    const int wave = tid >> 5;
    const int lane = tid & 31;
    const int l16  = lane & 15;
    const int half = lane >> 4;   // 0: lanes 0-15, 1: lanes 16-31
    const int n0   = wave * 16;   // this wave's output-column block

    // Stage input tile (coalesced); clamp rows for the (non-existent here) tail.
    for (int i = tid; i < 16 * PAD_K; i += 128) {
        int r = i / PAD_K;
        int c = i - r * PAD_K;
        int rr = m0 + r;
        rr = rr < n_nodes ? rr : n_nodes - 1;
        lds_in[i] = hin[(long long)rr * PAD_K + c];
    }
    __syncthreads();

    // ---- GEMM1: 16x16 tile of h = Hin(16x68) @ w1p(68x16-slice) ----
    // A layout (16x4 f32): v0 = A[l16][k0 + 2*half], v1 = A[l16][k0 + 2*half + 1]
    // B layout (4x16 f32): v0 = B[k0 + 2*half][n0+l16], v1 = B[k0+2*half+1][n0+l16]
    v8f acc = {};
    for (int k0 = 0; k0 < PAD_K; k0 += 4) {
        int ka = k0 + half * 2;
        v2f a, b;
        a.x = lds_in[l16 * PAD_K + ka];
        a.y = lds_in[l16 * PAD_K + ka + 1];
        b.x = w1p[ka * HID_C + n0 + l16];
        b.y = w1p[(ka + 1) * HID_C + n0 + l16];
        acc = __builtin_amdgcn_wmma_f32_16x16x4_f32(
            /*neg_a=*/false, a, /*neg_b=*/false, b,
            /*c_mod=*/(short)0, acc, /*reuse_a=*/false, /*reuse_b=*/false);
    }

    // bias + ReLU -> LDS h tile. D element i maps to row (i + 8*half), col n0+l16.
    float bias1 = b1[n0 + l16];
#pragma unroll
    for (int i = 0; i < 8; ++i) {
        float v = acc[i] + bias1;
        lds_h[(i + half * 8) * PAD_K + (n0 + l16)] = v > 0.0f ? v : 0.0f;
    }
    __syncthreads();

    // ---- GEMM2: 16x16 tile of out = h(16x64) @ w2(64x16-slice) ----
    v8f acc2 = {};
    for (int k0 = 0; k0 < HID_C; k0 += 4) {
        int ka = k0 + half * 2;
        v2f a, b;
        a.x = lds_h[l16 * PAD_K + ka];
        a.y = lds_h[l16 * PAD_K + ka + 1];
        b.x = w2[ka * HID_C + n0 + l16];
        b.y = w2[(ka + 1) * HID_C + n0 + l16];
        acc2 = __builtin_amdgcn_wmma_f32_16x16x4_f32(
            false, a, false, b, (short)0, acc2, false, false);
    }

    float bias2 = b2[n0 + l16];
#pragma unroll
    for (int i = 0; i < 8; ++i) {
        int orow = m0 + i + half * 8;
        if (orow < n_nodes)
            out[(long long)orow * HID_C + n0 + l16] = acc2[i] + bias2;
    }
}

// ---------------------------------------------------------------------------
extern "C" void kernel_launch(void* const* d_in, const int* in_sizes, int n_in,
                              void* d_out, int out_size, void* d_ws, size_t ws_size,
                              hipStream_t stream) {
    const float*     x          = (const float*)d_in[0];
    const long long* edge_index = (const long long*)d_in[1];  // int64 per reference
    const float*     w1         = (const float*)d_in[2];
    const float*     b1         = (const float*)d_in[3];
    const float*     w2         = (const float*)d_in[4];
    const float*     b2         = (const float*)d_in[5];
    float*           out        = (float*)d_out;

    const int n_nodes = in_sizes[0] / IN_C;
    const int n_edges = in_sizes[1] / 2;

    // Workspace layout: [w1p: 68*64 floats][hin: n_nodes*68 floats]
    float* w1p = (float*)d_ws;
    float* hin = w1p + PAD_K * HID_C;

    {   // Hin = x (padded)
        int total = n_nodes * PAD_K;
        gin_init_hin<<<(total + 255) / 256, 256, 0, stream>>>(x, hin, n_nodes);
    }
    gin_pad_w1<<<(PAD_K * HID_C + 255) / 256, 256, 0, stream>>>(w1, w1p);

    {   // edge scatter: one wave per edge
        long long total_threads = (long long)n_edges * 32;
        int blocks = (int)((total_threads + 255) / 256);
        gin_scatter<<<blocks, 256, 0, stream>>>(x, edge_index, hin, n_edges);
    }

    {   // fused 2-layer MLP on WMMA f32 path
        int blocks = (n_nodes + 15) / 16;
        gin_mlp_wmma<<<blocks, 128, 0, stream>>>(hin, w1p, b1, w2, b2, out, n_nodes);
    }
}